// SE_4123168604913
// MI455X (gfx1250) — compile-verified
//
#include <hip/hip_runtime.h>

typedef __attribute__((ext_vector_type(2))) float v2f;
typedef __attribute__((ext_vector_type(4))) float v4f;
typedef __attribute__((ext_vector_type(8))) float v8f;

#define SE_B   16
#define SE_C   256
#define SE_HW  (128 * 128)   // 16384 elements per (b,c) plane

// ---------------------------------------------------------------------------
// Kernel 1: global average pool.  One block per (b,c) plane (4096 blocks).
// 256 threads * 16 iterations of 128-bit loads = 16384 floats, coalesced.
// ---------------------------------------------------------------------------
__global__ __launch_bounds__(256) void se_pool(const float* __restrict__ U,
                                               float* __restrict__ z) {
  const int plane = blockIdx.x;                       // 0 .. 4095
  const v4f* p = (const v4f*)(U + (size_t)plane * SE_HW);
  float acc = 0.f;
#pragma unroll
  for (int i = 0; i < 16; ++i) {
    v4f v = p[threadIdx.x + i * 256];
    acc += (v.x + v.y) + (v.z + v.w);
  }
  // wave32 reduction
#pragma unroll
  for (int off = 16; off > 0; off >>= 1)
    acc += __shfl_down(acc, off, 32);
  __shared__ float ws[8];
  if ((threadIdx.x & 31) == 0) ws[threadIdx.x >> 5] = acc;
  __syncthreads();
  if (threadIdx.x == 0) {
    float t = 0.f;
#pragma unroll
    for (int w = 0; w < 8; ++w) t += ws[w];
    z[plane] = t * (1.0f / SE_HW);
  }
}

// ---------------------------------------------------------------------------
// Kernel 2: SE MLP on one wave using V_WMMA_F32_16X16X4_F32.
//   GEMM1: z2[16,128] = relu(z[16,256] x w_sq^T[256,128])     (8 N-tiles, K=256)
//   GEMM2: s [16,256] = sigmoid(z2[16,128] x w_ex^T[128,256]) (16 N-tiles, K=128)
// Batch=16 fills the WMMA M dimension exactly. EXEC is all-ones (32 uniform
// threads, no divergence around the WMMA ops).
// ---------------------------------------------------------------------------
__global__ __launch_bounds__(32) void se_mlp(const float* __restrict__ z,
                                             const float* __restrict__ w_sq,
                                             const float* __restrict__ w_ex,
                                             float* __restrict__ s_out) {
  __shared__ float zs[SE_B][SE_C];        // 16 KB
  __shared__ float z2[SE_B][SE_C / 2];    // 8 KB

  const int lane  = threadIdx.x;          // 0..31
  const int m     = lane & 15;            // A-row / B-col index
  const int khalf = lane >> 4;            // 0: K={0,1}, 1: K={2,3}

  // Stage z into LDS
  for (int i = lane; i < SE_B * SE_C; i += 32)
    zs[i / SE_C][i % SE_C] = z[i];
  __syncthreads();

  // ---- GEMM1 + ReLU -> z2 in LDS ----
  for (int nt = 0; nt < 8; ++nt) {
    v8f acc = {};
    for (int k0 = 0; k0 < SE_C; k0 += 4) {
      const int ka = k0 + 2 * khalf;
      v2f a  = { zs[m][ka], zs[m][ka + 1] };                 // A[m][k] = z[b=m][k]
      v2f bm = { w_sq[(nt * 16 + m) * SE_C + ka],            // B[k][n] = w_sq[n][k]
                 w_sq[(nt * 16 + m) * SE_C + ka + 1] };
      acc = __builtin_amdgcn_wmma_f32_16x16x4_f32(
          false, a, false, bm, (short)0, acc, false, false);
    }
#pragma unroll
    for (int v = 0; v < 8; ++v) {
      float val = acc[v] > 0.f ? acc[v] : 0.f;               // ReLU
      z2[v + 8 * khalf][nt * 16 + m] = val;                  // C layout -> LDS
    }
  }
  __syncthreads();

  // ---- GEMM2 + sigmoid -> s_out (global) ----
  for (int nt = 0; nt < 16; ++nt) {
    v8f acc = {};
    for (int k0 = 0; k0 < SE_C / 2; k0 += 4) {
      const int ka = k0 + 2 * khalf;
      v2f a  = { z2[m][ka], z2[m][ka + 1] };                 // A[m][k] = z2[b=m][k]
      v2f bm = { w_ex[(nt * 16 + m) * (SE_C / 2) + ka],      // B[k][n] = w_ex[n][k]
                 w_ex[(nt * 16 + m) * (SE_C / 2) + ka + 1] };
      acc = __builtin_amdgcn_wmma_f32_16x16x4_f32(
          false, a, false, bm, (short)0, acc, false, false);
    }
#pragma unroll
    for (int v = 0; v < 8; ++v) {
      float g = 1.0f / (1.0f + __expf(-acc[v]));             // sigmoid
      s_out[(v + 8 * khalf) * SE_C + nt * 16 + m] = g;
    }
  }
}

// ---------------------------------------------------------------------------
// Kernel 3: out = U * s[b,c].  Planes traversed in REVERSE order so the tail
// of kernel 1's U-stream (still resident in the 192 MB L2) is re-hit instead
// of LRU-thrashed. Output written with non-temporal stores so the 256 MB
// write stream does not evict the U lines.
// ---------------------------------------------------------------------------
__global__ __launch_bounds__(256) void se_scale(const float* __restrict__ U,
                                                const float* __restrict__ s,
                                                float* __restrict__ out) {
  const int plane = (int)gridDim.x - 1 - (int)blockIdx.x;    // reversed
  const float g = s[plane];                                  // block-uniform -> s_load
  const v4f* p = (const v4f*)(U + (size_t)plane * SE_HW);
  v4f* q = (v4f*)(out + (size_t)plane * SE_HW);
#pragma unroll
  for (int i = 15; i >= 0; --i) {
    const int idx = threadIdx.x + i * 256;
    v4f v = p[idx];                                          // RT load (L2 reuse)
    v *= g;
    __builtin_nontemporal_store(v, &q[idx]);                 // NT store
  }
}

// ---------------------------------------------------------------------------
extern "C" void kernel_launch(void* const* d_in, const int* in_sizes, int n_in,
                              void* d_out, int out_size, void* d_ws, size_t ws_size,
                              hipStream_t stream) {
  const float* U    = (const float*)d_in[0];   // [16,256,128,128]
  const float* w_sq = (const float*)d_in[1];   // [128,256]
  const float* w_ex = (const float*)d_in[2];   // [256,128]
  float* out = (float*)d_out;

  float* z = (float*)d_ws;                     // [16,256]  pooled means
  float* s = z + SE_B * SE_C;                  // [16,256]  gates

  se_pool <<<SE_B * SE_C, 256, 0, stream>>>(U, z);
  se_mlp  <<<1, 32, 0, stream>>>(z, w_sq, w_ex, s);
  se_scale<<<SE_B * SE_C, 256, 0, stream>>>(U, s, out);
}